// MentalNetNaive_58737972740329
// MI455X (gfx1250) — compile-verified
//
#include <hip/hip_runtime.h>
#include <hip/hip_bf16.h>

typedef __attribute__((ext_vector_type(2))) float v2f;
typedef __attribute__((ext_vector_type(8))) float v8f;

#define HDIM 64
#define NEG_SLOPE 0.2f

// ---------------- WMMA GEMM: Hout[N,64] = X[N,Din] @ W[Din,64] (fp32) ----------------
// One wave per 16-row tile; 4 column tiles of 16; K stepped by 4 using
// V_WMMA_F32_16X16X4_F32. A-frag (16x4): lanes 0-15 hold K=k..k+1, lanes 16-31 K=k+2..k+3.
// B-frag (4x16): same K split across lane halves, N striped across lanes.
// C/D (16x16): VGPR j -> row j + 8*(lane/16), col = lane%16.
__global__ void gemm_wmma(const float* __restrict__ X, const float* __restrict__ W,
                          float* __restrict__ Hout, int nrows, int Din) {
    const int wave = threadIdx.x >> 5;
    const int lane = threadIdx.x & 31;
    const int rowTile = blockIdx.x * 4 + wave;
    const int rowBase = rowTile * 16;
    if (rowBase >= nrows) return;               // wave-uniform exit; EXEC stays all-1s
    const int half = lane >> 4;                 // 0: K pair 0-1, 1: K pair 2-3
    const int lidx = lane & 15;                 // M (for A) / N (for B,C)

    v8f acc0 = {}, acc1 = {}, acc2 = {}, acc3 = {};
    for (int k = 0; k < Din; k += 4) {
        const int ka = k + half * 2;
        const float* xp = X + (size_t)(rowBase + lidx) * Din + ka;
        v2f a; a.x = xp[0]; a.y = xp[1];

        const float* wp = W + (size_t)ka * HDIM + lidx;
        v2f b0; b0.x = wp[0];      b0.y = wp[HDIM];
        v2f b1; b1.x = wp[16];     b1.y = wp[HDIM + 16];
        v2f b2; b2.x = wp[32];     b2.y = wp[HDIM + 32];
        v2f b3; b3.x = wp[48];     b3.y = wp[HDIM + 48];

        acc0 = __builtin_amdgcn_wmma_f32_16x16x4_f32(false, a, false, b0, (short)0, acc0, false, false);
        acc1 = __builtin_amdgcn_wmma_f32_16x16x4_f32(false, a, false, b1, (short)0, acc1, false, false);
        acc2 = __builtin_amdgcn_wmma_f32_16x16x4_f32(false, a, false, b2, (short)0, acc2, false, false);
        acc3 = __builtin_amdgcn_wmma_f32_16x16x4_f32(false, a, false, b3, (short)0, acc3, false, false);
    }
#pragma unroll
    for (int j = 0; j < 8; ++j) {
        float* op = Hout + (size_t)(rowBase + j + 8 * half) * HDIM + lidx;
        op[0]  = acc0[j];
        op[16] = acc1[j];
        op[32] = acc2[j];
        op[48] = acc3[j];
    }
}

// ---------------- per-node attention logits: alpha_s[n] = h[n].a_s, alpha_d[n] = h[n].a_d --
__global__ void alpha_kernel(const float* __restrict__ H, const float* __restrict__ avs,
                             const float* __restrict__ avd, float* __restrict__ als,
                             float* __restrict__ ald, int n) {
    const int wave = threadIdx.x >> 5;
    const int lane = threadIdx.x & 31;
    const int node = blockIdx.x * 8 + wave;
    if (node >= n) return;
    const float* h = H + (size_t)node * HDIM;
    float s0 = h[lane] * avs[lane] + h[lane + 32] * avs[lane + 32];
    float s1 = h[lane] * avd[lane] + h[lane + 32] * avd[lane + 32];
#pragma unroll
    for (int off = 16; off > 0; off >>= 1) {
        s0 += __shfl_down(s0, off, 32);
        s1 += __shfl_down(s1, off, 32);
    }
    if (lane == 0) { als[node] = s0; ald[node] = s1; }
}

// ---------------- init: out rows = bias, segment-max key = 0 (== -inf), segment-sum = 0 ----
__global__ void init_kernel(float* __restrict__ out, const float* __restrict__ bias,
                            unsigned* __restrict__ mkey, float* __restrict__ ssum, int n) {
    const int i = blockIdx.x * 256 + threadIdx.x;
    const int node = i >> 6, f = i & 63;
    if (node >= n) return;
    out[i] = bias[f];
    if (f == 0) { mkey[node] = 0u; ssum[node] = 0.f; }
}

__device__ __forceinline__ unsigned fkey(float f) {
    unsigned b = __float_as_uint(f);
    return (b & 0x80000000u) ? ~b : (b | 0x80000000u);
}
__device__ __forceinline__ float fdecode(unsigned k) {
    unsigned b = (k & 0x80000000u) ? (k & 0x7FFFFFFFu) : ~k;
    return __uint_as_float(b);
}

// ---------------- pass A: e = LeakyReLU(as[src]+ad[dst]); segment max over dst ------------
__global__ void edgeA(const int* __restrict__ srcA, const int* __restrict__ dstA,
                      const float* __restrict__ als, const float* __restrict__ ald,
                      float* __restrict__ ebuf, unsigned* __restrict__ mkey,
                      int E, int Etot) {
    const int e = blockIdx.x * 256 + threadIdx.x;
    if (e >= Etot) return;
    int s, d;
    if (e < E) { s = srcA[e]; d = dstA[e]; } else { s = d = e - E; }  // self loops
    float v = als[s] + ald[d];
    v = (v >= 0.f) ? v : NEG_SLOPE * v;
    ebuf[e] = v;
    atomicMax(mkey + d, fkey(v));
}

// ---------------- pass B: p = exp(e - max[dst]); segment sum over dst ---------------------
__global__ void edgeB(const int* __restrict__ dstA, const unsigned* __restrict__ mkey,
                      float* __restrict__ ebuf, float* __restrict__ ssum, int E, int Etot) {
    const int e = blockIdx.x * 256 + threadIdx.x;
    if (e >= Etot) return;
    const int d = (e < E) ? dstA[e] : (e - E);
    float p = __expf(ebuf[e] - fdecode(mkey[d]));
    ebuf[e] = p;
    atomicAdd(ssum + d, p);
}

// ---------------- pass C: out[dst] += (p/(s+eps))*ew * h[src] (wave per edge) -------------
__global__ void edgeC(const int* __restrict__ srcA, const int* __restrict__ dstA,
                      const float* __restrict__ ewr, const float* __restrict__ ebuf,
                      const float* __restrict__ ssum, const float* __restrict__ H,
                      float* __restrict__ out, int E, int Etot) {
    const int wave = threadIdx.x >> 5;
    const int lane = threadIdx.x & 31;
    const int e = blockIdx.x * 8 + wave;
    if (e >= Etot) return;
    int s, d; float w;
    if (e < E) { s = srcA[e]; d = dstA[e]; w = ewr[e]; } else { s = d = e - E; w = 1.f; }
    const float coef = ebuf[e] / (ssum[d] + 1e-16f) * w;
    const float* hs = H + (size_t)s * HDIM;
    float* od = out + (size_t)d * HDIM;
    atomicAdd(od + lane,       coef * hs[lane]);
    atomicAdd(od + lane + 32,  coef * hs[lane + 32]);
}

__global__ void relu_kernel(float* __restrict__ h, int total) {
    const int i = blockIdx.x * 256 + threadIdx.x;
    if (i < total) h[i] = fmaxf(h[i], 0.f);
}

// write relation r's block into [N, 192] output
__global__ void copy_out(const float* __restrict__ h, float* __restrict__ out, int r, int n) {
    const int i = blockIdx.x * 256 + threadIdx.x;
    const int node = i >> 6, f = i & 63;
    if (node >= n) return;
    out[(size_t)node * 192 + r * 64 + f] = h[i];
}

extern "C" void kernel_launch(void* const* d_in, const int* in_sizes, int n_in,
                              void* d_out, int out_size, void* d_ws, size_t ws_size,
                              hipStream_t stream) {
    const float* x     = (const float*)d_in[0];
    const float* ew    = (const float*)d_in[1];
    const float* W0    = (const float*)d_in[2];
    const float* W1    = (const float*)d_in[3];
    const float* W2    = (const float*)d_in[4];
    const float* a_src = (const float*)d_in[5];
    const float* a_dst = (const float*)d_in[6];
    const float* bias  = (const float*)d_in[7];
    const int*   eidx  = (const int*)d_in[8];
    float* outp = (float*)d_out;

    const int D = 128, H = 64, R = 3;
    const int N = in_sizes[0] / D;
    const int E = in_sizes[1] / R;
    const int Etot = E + N;

    // workspace layout (floats): hA[N*64] hB[N*64] als[N] ald[N] mkey[N] ssum[N] ebuf[Etot]
    float* hA = (float*)d_ws;
    float* hB = hA + (size_t)N * H;
    float* als = hB + (size_t)N * H;
    float* ald = als + N;
    unsigned* mkey = (unsigned*)(ald + N);
    float* ssum = (float*)(mkey + N);
    float* ebuf = ssum + N;

    const int gemmGrid   = ((N + 15) / 16 + 3) / 4;   // 4 waves (16-row tiles) per block
    const int nodeWGrid  = (N + 7) / 8;               // 8 waves per block
    const int nhGrid     = (int)(((size_t)N * H + 255) / 256);
    const int eGrid      = (Etot + 255) / 256;
    const int eWaveGrid  = (Etot + 7) / 8;

    for (int r = 0; r < R; ++r) {
        const int*   srcA = eidx + (size_t)(2 * r + 0) * E;
        const int*   dstA = eidx + (size_t)(2 * r + 1) * E;
        const float* ewr  = ew + (size_t)r * E;
        for (int i = 0; i < 3; ++i) {
            const float* Wm  = (i == 0) ? (W0 + (size_t)r * D * H)
                             : (i == 1) ? (W1 + (size_t)r * H * H)
                                        : (W2 + (size_t)r * H * H);
            const int    Din = (i == 0) ? D : H;
            const float* in  = (i == 0) ? x : hB;
            const float* avs = a_src + (size_t)(r * 3 + i) * H;
            const float* avd = a_dst + (size_t)(r * 3 + i) * H;
            const float* bi  = bias  + (size_t)(r * 3 + i) * H;

            gemm_wmma  <<<gemmGrid,  128, 0, stream>>>(in, Wm, hA, N, Din);
            alpha_kernel<<<nodeWGrid, 256, 0, stream>>>(hA, avs, avd, als, ald, N);
            init_kernel<<<nhGrid,    256, 0, stream>>>(hB, bi, mkey, ssum, N);
            edgeA      <<<eGrid,     256, 0, stream>>>(srcA, dstA, als, ald, ebuf, mkey, E, Etot);
            edgeB      <<<eGrid,     256, 0, stream>>>(dstA, mkey, ebuf, ssum, E, Etot);
            edgeC      <<<eWaveGrid, 256, 0, stream>>>(srcA, dstA, ewr, ebuf, ssum, hA, hB, E, Etot);
            if (i != 2) relu_kernel<<<nhGrid, 256, 0, stream>>>(hB, N * H);
            else        copy_out   <<<nhGrid, 256, 0, stream>>>(hB, outp, r, N);
        }
    }
}